// TopologyAwareFocalLoss_26061861552441
// MI455X (gfx1250) — compile-verified
//
#include <hip/hip_runtime.h>

#define DIM_H 96
#define DIM_W 96
#define DIM_D 96
#define NVOL 4
#define VOXELS (DIM_H * DIM_W * DIM_D)   // 884736
#define NTOT   (NVOL * VOXELS)           // 3538944

#define THREADS 256
#define FOCAL_BLOCKS 1024

// workspace layout (bytes)
#define PART_OFF   0          // float partials[FOCAL_BLOCKS]  (4 KB)
#define CC_OFF     4096       // unsigned cc[8]                (32 B)
#define PARENT_OFF 8192       // int parent[NTOT]              (~13.5 MB)

typedef float v2f __attribute__((ext_vector_type(2)));
typedef float v8f __attribute__((ext_vector_type(8)));

// ---------------- focal loss ----------------

__device__ __forceinline__ float focal_term(float x, float t) {
    // p = sigmoid(x); bce = softplus(x) - x*t
    float p  = 1.0f / (1.0f + expf(-x));
    bool pos = (t > 0.5f);
    float pt = pos ? p : (1.0f - p);
    float sp = fmaxf(x, 0.0f) + log1pf(expf(-fabsf(x)));
    float bce = sp - x * t;
    float alpha = pos ? 0.25f : 0.75f;
    float om = 1.0f - pt;
    return alpha * om * om * bce;        // GAMMA = 2
}

__global__ void focal_partial_kernel(const float* __restrict__ x,
                                     const float* __restrict__ t,
                                     float* __restrict__ partials, int n) {
    float s = 0.0f;
    for (int i = blockIdx.x * blockDim.x + threadIdx.x; i < n;
         i += gridDim.x * blockDim.x) {
        s += focal_term(x[i], t[i]);
    }
    // Exact f32 wave-level sum via V_WMMA_F32_16X16X4_F32:
    // A[m][k] holds the per-lane partials (lane L<16 -> A[L][0], lane L>=16 -> A[L-16][2]),
    // B = ones(4x16)  =>  D[m][n] = rowsum(A)[m] = s[m] + s[m+16].
    v2f a; a.x = s;    a.y = 0.0f;
    v2f b; b.x = 1.0f; b.y = 1.0f;
    v8f c = {};
    c = __builtin_amdgcn_wmma_f32_16x16x4_f32(false, a, false, b,
                                              (short)0, c, false, false);
    float wsum = c[0] + c[1] + c[2] + c[3] + c[4] + c[5] + c[6] + c[7];
    wsum += __shfl_xor(wsum, 16, 32);    // combine halves -> full 32-lane sum

    __shared__ float lds[THREADS / 32];
    int wave = threadIdx.x >> 5;
    int lane = threadIdx.x & 31;
    if (lane == 0) lds[wave] = wsum;
    __syncthreads();
    if (threadIdx.x == 0) {
        float bs = 0.0f;
        for (int wv = 0; wv < THREADS / 32; ++wv) bs += lds[wv];  // fixed order
        partials[blockIdx.x] = bs;
    }
}

// ---------------- connected components (union-find) ----------------

__device__ __forceinline__ bool fg_of(const float* __restrict__ src, int i, int mode) {
    float v = src[i];
    return (mode == 0) ? (v > 0.0f)      // sigmoid(x) > 0.5  <=>  x > 0
                       : (v > 0.5f);     // target > 0.5
}

__device__ __forceinline__ int uf_find(int* P, int i) {
    int p = __hip_atomic_load(&P[i], __ATOMIC_RELAXED, __HIP_MEMORY_SCOPE_AGENT);
    while (p != i) {
        i = p;
        p = __hip_atomic_load(&P[i], __ATOMIC_RELAXED, __HIP_MEMORY_SCOPE_AGENT);
    }
    return i;
}

__device__ __forceinline__ void uf_union(int* P, int a, int b) {
    bool done = false;
    while (!done) {
        a = uf_find(P, a);
        b = uf_find(P, b);
        if (a < b) {
            int old = atomicMin(&P[b], a);
            done = (old == b);
            b = old;
        } else if (b < a) {
            int old = atomicMin(&P[a], b);
            done = (old == a);
            a = old;
        } else {
            done = true;
        }
    }
}

__global__ void init_parent_kernel(int* __restrict__ P, int n) {
    int i = blockIdx.x * blockDim.x + threadIdx.x;
    if (i < n) P[i] = i;
}

__global__ void merge_kernel(const float* __restrict__ src, int* P, int mode) {
    int i = blockIdx.x * blockDim.x + threadIdx.x;
    if (i >= NTOT) return;
    if (!fg_of(src, i, mode)) return;
    int cidx = i % VOXELS;
    int dd  = cidx % DIM_D;
    int rem = cidx / DIM_D;
    int ww  = rem % DIM_W;
    int hh  = rem / DIM_W;
    if (dd + 1 < DIM_D && fg_of(src, i + 1, mode))             uf_union(P, i, i + 1);
    if (ww + 1 < DIM_W && fg_of(src, i + DIM_D, mode))         uf_union(P, i, i + DIM_D);
    if (hh + 1 < DIM_H && fg_of(src, i + DIM_W * DIM_D, mode)) uf_union(P, i, i + DIM_W * DIM_D);
}

__global__ void count_kernel(const float* __restrict__ src, const int* __restrict__ P,
                             unsigned* __restrict__ cc, int mode, int ccbase) {
    int i = blockIdx.x * blockDim.x + threadIdx.x;
    if (i >= NTOT) return;
    if (!fg_of(src, i, mode)) return;
    if (P[i] == i) {                       // root: min index of its component
        int vol = i / VOXELS;
        atomicAdd(&cc[ccbase + vol], 1u);  // integer atomics -> exact & deterministic
    }
}

// ---------------- finalize ----------------

__global__ void zero_counters_kernel(unsigned* cc) {
    if (threadIdx.x < 8) cc[threadIdx.x] = 0u;
}

__global__ void finalize_kernel(const float* __restrict__ partials,
                                const unsigned* __restrict__ cc,
                                float* __restrict__ out, int nPart) {
    if (threadIdx.x != 0 || blockIdx.x != 0) return;
    float s = 0.0f;
    for (int i = 0; i < nPart; ++i) s += partials[i];   // fixed sequential order
    float focal = s / (float)NTOT;

    // volumes flattened as v = b*2 + c  (B=2, C=2); batch-mean per class
    float topo = 0.0f;
    for (int c = 0; c < 2; ++c) {
        float pp = 0.5f * ((float)cc[c]     + (float)cc[2 + c]);
        float tt = 0.5f * ((float)cc[4 + c] + (float)cc[6 + c]);
        topo += fabsf(pp - tt);
    }
    topo *= 0.5f;                   // mean over classes
    out[0] = focal + 0.1f * topo;   // LAMBDA_T = 0.1
}

// ---------------- launch ----------------

extern "C" void kernel_launch(void* const* d_in, const int* in_sizes, int n_in,
                              void* d_out, int out_size, void* d_ws, size_t ws_size,
                              hipStream_t stream) {
    (void)n_in; (void)out_size; (void)ws_size;
    const float* logits = (const float*)d_in[0];
    const float* targ   = (const float*)d_in[1];
    float* out = (float*)d_out;

    char* ws = (char*)d_ws;
    float*    partials = (float*)(ws + PART_OFF);
    unsigned* cc       = (unsigned*)(ws + CC_OFF);
    int*      parent   = (int*)(ws + PARENT_OFF);

    int n = in_sizes[0];  // 3538944
    int blocks = (NTOT + THREADS - 1) / THREADS;

    zero_counters_kernel<<<1, 32, 0, stream>>>(cc);
    focal_partial_kernel<<<FOCAL_BLOCKS, THREADS, 0, stream>>>(logits, targ, partials, n);

    // pred components (fg = logits > 0)
    init_parent_kernel<<<blocks, THREADS, 0, stream>>>(parent, NTOT);
    merge_kernel<<<blocks, THREADS, 0, stream>>>(logits, parent, 0);
    count_kernel<<<blocks, THREADS, 0, stream>>>(logits, parent, cc, 0, 0);

    // target components (fg = target > 0.5), reuse parent array
    init_parent_kernel<<<blocks, THREADS, 0, stream>>>(parent, NTOT);
    merge_kernel<<<blocks, THREADS, 0, stream>>>(targ, parent, 1);
    count_kernel<<<blocks, THREADS, 0, stream>>>(targ, parent, cc, 1, 4);

    finalize_kernel<<<1, 1, 0, stream>>>(partials, cc, out, FOCAL_BLOCKS);
}